// EVSSM_33500744909102
// MI455X (gfx1250) — compile-verified
//
#include <hip/hip_runtime.h>
#include <hip/hip_bf16.h>
#include <math.h>

// ---------------- problem constants ----------------
#define L_TOT   4096
#define HH      64
#define WWID    64
#define C_IN    96
#define DI      192
#define NS      16
#define RK      6
#define KDIR    4
#define XC      38      // RK + 2*NS
#define HFD     255
#define HFDP    256     // padded ld for pout GEMM operands
#define CHUNKS  32
#define CLEN    128     // L_TOT / CHUNKS

typedef float v2f __attribute__((ext_vector_type(2)));
typedef float v8f __attribute__((ext_vector_type(8)));

__device__ __forceinline__ float sig_(float x)  { return 1.f / (1.f + __expf(-x)); }
__device__ __forceinline__ float silu_(float x) { return x * sig_(x); }
__device__ __forceinline__ int   imin_(int a, int b) { return a < b ? a : b; }

// =====================================================================
// fp32 WMMA GEMM:  C[M,N] = A[M,K] @ W[N,K]^T   (weights stored (out,in))
// One wave32 computes a 16(M) x 32(N) tile: two V_WMMA_F32_16X16X4_F32
// accumulators sharing one A fragment. 4 waves / block.
// Main K loop is branchless (aligned v2f loads); ragged K handled by one
// clamped+cndmask tail step; ragged N by clamped B pointers + store guard.
// Batched over blockIdx.z. EPI==1: per-column bias + softplus epilogue.
// =====================================================================
template <int EPI>
__global__ void gemm_wmma(const float* __restrict__ A, const float* __restrict__ W,
                          float* __restrict__ C, const float* __restrict__ bias,
                          int M, int N, int K, int lda, int ldw, int ldc,
                          int sA, int sW, int sC, int sB)
{
    const int bz = blockIdx.z;
    A += (size_t)bz * sA;
    W += (size_t)bz * sW;
    C += (size_t)bz * sC;
    const float* bvec = bias ? (bias + (size_t)bz * sB) : nullptr;

    const int wave  = threadIdx.x >> 5;
    const int lane  = threadIdx.x & 31;
    const int ntile = blockIdx.y * 4 + wave;          // 32 output cols per wave
    if (ntile * 32 >= N) return;                      // wave-uniform exit
    const int m0   = blockIdx.x * 16;
    const int n0   = ntile * 32;
    const int half = lane >> 4;                       // K-pair / M-row-half select
    const int r    = lane & 15;

    const int nn0 = n0 + r;
    const int nn1 = n0 + 16 + r;
    const int nc0 = imin_(nn0, N - 1);                // clamped: garbage cols discarded at store
    const int nc1 = imin_(nn1, N - 1);

    const float* __restrict__ Arow = A + (size_t)(m0 + r) * lda + 2 * half;
    const float* __restrict__ W0   = W + (size_t)nc0 * ldw + 2 * half;
    const float* __restrict__ W1   = W + (size_t)nc1 * ldw + 2 * half;

    v8f acc0 = {}, acc1 = {};
    const int K4 = K & ~3;
    for (int k0 = 0; k0 < K4; k0 += 4) {
        v2f a  = *(const v2f*)(Arow + k0);            // 64-bit aligned loads
        v2f b0 = *(const v2f*)(W0 + k0);
        v2f b1 = *(const v2f*)(W1 + k0);
        acc0 = __builtin_amdgcn_wmma_f32_16x16x4_f32(false, a, false, b0,
                                                     (short)0, acc0, false, false);
        acc1 = __builtin_amdgcn_wmma_f32_16x16x4_f32(false, a, false, b1,
                                                     (short)0, acc1, false, false);
    }
    if (K4 < K) {                                     // single ragged-K step, branchless
        const int ka  = K4 + 2 * half;
        const int kc0 = imin_(ka,     K - 1) - 2 * half - K4;
        const int kc1 = imin_(ka + 1, K - 1) - 2 * half - K4;
        v2f a, b0, b1;
        a.x  = (ka     < K) ? Arow[K4 + kc0] : 0.f;
        a.y  = (ka + 1 < K) ? Arow[K4 + kc1] : 0.f;
        b0.x = (ka     < K) ? W0[K4 + kc0]   : 0.f;
        b0.y = (ka + 1 < K) ? W0[K4 + kc1]   : 0.f;
        b1.x = (ka     < K) ? W1[K4 + kc0]   : 0.f;
        b1.y = (ka + 1 < K) ? W1[K4 + kc1]   : 0.f;
        acc0 = __builtin_amdgcn_wmma_f32_16x16x4_f32(false, a, false, b0,
                                                     (short)0, acc0, false, false);
        acc1 = __builtin_amdgcn_wmma_f32_16x16x4_f32(false, a, false, b1,
                                                     (short)0, acc1, false, false);
    }

#pragma unroll
    for (int j = 0; j < 8; ++j) {
        const int m = m0 + j + 8 * half;
        float v0 = acc0[j], v1 = acc1[j];
        if (EPI == 1) {
            v0 += bvec[nc0];
            v1 += bvec[nc1];
            v0 = (v0 > 20.f) ? v0 : log1pf(__expf(v0));
            v1 = (v1 > 20.f) ? v1 : log1pf(__expf(v1));
        }
        if (nn0 < N) C[(size_t)m * ldc + nn0] = v0;
        if (nn1 < N) C[(size_t)m * ldc + nn1] = v1;
    }
}

// =====================================================================
// RMS-LN (mean-square only, eps 1e-6): NCHW input -> pixel-major (L,C)
// =====================================================================
__global__ void rmsln_nchw(const float* __restrict__ x, const float* __restrict__ w,
                           const float* __restrict__ b, float* __restrict__ out, int C)
{
    int p = blockIdx.x * blockDim.x + threadIdx.x;
    if (p >= L_TOT) return;
    float s = 0.f;
    for (int c = 0; c < C; ++c) { float v = x[c * L_TOT + p]; s += v * v; }
    float r = rsqrtf(s / (float)C + 1e-6f);
    for (int c = 0; c < C; ++c)
        out[(size_t)p * C + c] = x[c * L_TOT + p] * r * w[c] + b[c];
}

// pixel-major (L,C) input variant
__global__ void rmsln_lc(const float* __restrict__ x, const float* __restrict__ w,
                         const float* __restrict__ b, float* __restrict__ out, int C)
{
    int p = blockIdx.x * blockDim.x + threadIdx.x;
    if (p >= L_TOT) return;
    const float* row = x + (size_t)p * C;
    float s = 0.f;
    for (int c = 0; c < C; ++c) s += row[c] * row[c];
    float r = rsqrtf(s / (float)C + 1e-6f);
    for (int c = 0; c < C; ++c) out[(size_t)p * C + c] = row[c] * r * w[c] + b[c];
}

// =====================================================================
// depthwise 3x3 SAME conv + bias + SiLU on xz[:, :DI]; scatter into the
// 4 scan directions (hw, wh, rev-hw, rev-wh), each pixel-major (L,DI)
// =====================================================================
__global__ void conv_silu_scatter(const float* __restrict__ xz, const float* __restrict__ cw,
                                  const float* __restrict__ cb, float* __restrict__ xs)
{
    int t = blockIdx.x * blockDim.x + threadIdx.x;
    if (t >= L_TOT * DI) return;
    int d = t % DI, p = t / DI;
    int h = p >> 6, w = p & 63;
    float acc = cb[d];
#pragma unroll
    for (int i = -1; i <= 1; ++i)
#pragma unroll
        for (int j = -1; j <= 1; ++j) {
            int hh = h + i, ww = w + j;
            if (hh >= 0 && hh < HH && ww >= 0 && ww < WWID)
                acc += xz[(size_t)(hh * WWID + ww) * (2 * DI) + d] * cw[d * 9 + (i + 1) * 3 + (j + 1)];
        }
    float v = silu_(acc);
    int pwh = (p & 63) * HH + (p >> 6);           // (w,h) flat index
    xs[(size_t)p * DI + d]                               = v;  // dir 0: hw
    xs[(size_t)(L_TOT + pwh) * DI + d]                   = v;  // dir 1: wh
    xs[(size_t)(2 * L_TOT + (L_TOT - 1 - p)) * DI + d]   = v;  // dir 2
    xs[(size_t)(3 * L_TOT + (L_TOT - 1 - pwh)) * DI + d] = v;  // dir 3
}

// =====================================================================
// Selective scan, two-level. h_l = a_l h_{l-1} + b_l,
// a = exp(delta*A), b = delta*u*Bs.
// =====================================================================
__global__ void scan_phaseA(const float* __restrict__ delta, const float* __restrict__ xs,
                            const float* __restrict__ xdbl, const float* __restrict__ A_logs,
                            float* __restrict__ agg)
{
    int t = blockIdx.x * blockDim.x + threadIdx.x;
    if (t >= KDIR * CHUNKS * NS * DI) return;
    int d = t % DI;  int r = t / DI;
    int n = r % NS;  int r2 = r / NS;
    int chunk = r2 % CHUNKS;  int k = r2 / CHUNKS;
    int kd = k * DI + d;
    float Ac = -__expf(A_logs[kd * NS + n]);
    const float* dk = delta + (size_t)k * L_TOT * DI;
    const float* uk = xs    + (size_t)k * L_TOT * DI;
    const float* bk = xdbl  + (size_t)k * L_TOT * XC + RK + n;
    float P = 1.f, S = 0.f;
    int l0 = chunk * CLEN;
    for (int i = 0; i < CLEN; ++i) {
        int l = l0 + i;
        float dt = dk[(size_t)l * DI + d];
        float a  = __expf(dt * Ac);
        float b  = dt * uk[(size_t)l * DI + d] * bk[(size_t)l * XC];
        P *= a;
        S = a * S + b;
    }
    size_t idx = ((size_t)(kd * NS + n)) * CHUNKS + chunk;
    agg[2 * idx] = P; agg[2 * idx + 1] = S;
}

__global__ void scan_phaseB(const float* __restrict__ agg, float* __restrict__ hIn)
{
    int t = blockIdx.x * blockDim.x + threadIdx.x;   // (k*DI+d)*NS+n
    if (t >= KDIR * DI * NS) return;
    float h = 0.f;
    for (int c = 0; c < CHUNKS; ++c) {
        hIn[(size_t)t * CHUNKS + c] = h;
        size_t idx = (size_t)t * CHUNKS + c;
        h = agg[2 * idx] * h + agg[2 * idx + 1];
    }
}

__global__ void scan_phaseC(const float* __restrict__ delta, const float* __restrict__ xs,
                            const float* __restrict__ xdbl, const float* __restrict__ A_logs,
                            const float* __restrict__ Ds, const float* __restrict__ hIn,
                            float* __restrict__ ys)
{
    int t = blockIdx.x * blockDim.x + threadIdx.x;
    if (t >= KDIR * CHUNKS * DI) return;
    int d = t % DI;  int r = t / DI;
    int chunk = r % CHUNKS;  int k = r / CHUNKS;
    int kd = k * DI + d;
    float Ac[NS], h[NS];
#pragma unroll
    for (int n = 0; n < NS; ++n) {
        Ac[n] = -__expf(A_logs[kd * NS + n]);
        h[n]  = hIn[(size_t)(kd * NS + n) * CHUNKS + chunk];
    }
    float Dp = Ds[kd];
    const float* dk = delta + (size_t)k * L_TOT * DI;
    const float* uk = xs    + (size_t)k * L_TOT * DI;
    const float* xb = xdbl  + (size_t)k * L_TOT * XC;
    float*       yk = ys    + (size_t)k * L_TOT * DI;
    int l0 = chunk * CLEN;
    for (int i = 0; i < CLEN; ++i) {
        int l = l0 + i;
        float dt = dk[(size_t)l * DI + d];
        float u  = uk[(size_t)l * DI + d];
        float du = dt * u;
        const float* row = xb + (size_t)l * XC;
        float y = 0.f;
#pragma unroll
        for (int n = 0; n < NS; ++n) {
            float a = __expf(dt * Ac[n]);
            h[n] = a * h[n] + du * row[RK + n];
            y += h[n] * row[RK + NS + n];
        }
        yk[(size_t)l * DI + d] = y + Dp * u;
    }
}

// =====================================================================
// Merge 4 directions, LayerNorm (mean+var, eps 1e-5), SiLU(z) gate
// =====================================================================
__global__ void combine_ln_gate(const float* __restrict__ ys, const float* __restrict__ xz,
                                const float* __restrict__ onw, const float* __restrict__ onb,
                                float* __restrict__ yn)
{
    int p = blockIdx.x * blockDim.x + threadIdx.x;
    if (p >= L_TOT) return;
    int pwh = (p & 63) * HH + (p >> 6);
    const float* y0 = ys;
    const float* y1 = ys + (size_t)L_TOT * DI;
    const float* y2 = ys + (size_t)2 * L_TOT * DI;
    const float* y3 = ys + (size_t)3 * L_TOT * DI;
    float mu = 0.f, s2 = 0.f;
    for (int d = 0; d < DI; ++d) {
        float v = y0[(size_t)p * DI + d]
                + y2[(size_t)(L_TOT - 1 - p) * DI + d]
                + y1[(size_t)pwh * DI + d]
                + y3[(size_t)(L_TOT - 1 - pwh) * DI + d];
        mu += v; s2 += v * v;
    }
    mu /= (float)DI;
    float var  = s2 / (float)DI - mu * mu;
    float rinv = rsqrtf(var + 1e-5f);
    for (int d = 0; d < DI; ++d) {
        float v = y0[(size_t)p * DI + d]
                + y2[(size_t)(L_TOT - 1 - p) * DI + d]
                + y1[(size_t)pwh * DI + d]
                + y3[(size_t)(L_TOT - 1 - pwh) * DI + d];
        float z = xz[(size_t)p * (2 * DI) + DI + d];
        yn[(size_t)p * DI + d] = ((v - mu) * rinv * onw[d] + onb[d]) * silu_(z);
    }
}

// residual: x2(L,C) = x(NCHW) + attn(L,C)
__global__ void resid_add(const float* __restrict__ x, const float* __restrict__ attn,
                          float* __restrict__ x2)
{
    int t = blockIdx.x * blockDim.x + threadIdx.x;
    if (t >= L_TOT * C_IN) return;
    int c = t % C_IN, p = t / C_IN;
    x2[t] = x[c * L_TOT + p] + attn[t];
}

// EDFFN depthwise 3x3 (no bias) on hbuf(L,510), then gelu(h1)*h2 -> g(L,HFDP)
__global__ void dwconv_gelugate(const float* __restrict__ hbuf, const float* __restrict__ dww,
                                float* __restrict__ g)
{
    int t = blockIdx.x * blockDim.x + threadIdx.x;
    if (t >= L_TOT * HFD) return;
    int o = t % HFD, p = t / HFD;
    int h = p >> 6, w = p & 63;
    float a1 = 0.f, a2 = 0.f;
#pragma unroll
    for (int i = -1; i <= 1; ++i)
#pragma unroll
        for (int j = -1; j <= 1; ++j) {
            int hh = h + i, ww = w + j;
            if (hh >= 0 && hh < HH && ww >= 0 && ww < WWID) {
                const float* px = hbuf + (size_t)(hh * WWID + ww) * (2 * HFD);
                int widx = (i + 1) * 3 + (j + 1);
                a1 += px[o]       * dww[o * 9 + widx];
                a2 += px[o + HFD] * dww[(o + HFD) * 9 + widx];
            }
        }
    float ge = 0.5f * a1 * (1.f + erff(a1 * 0.70710678118654752f));
    g[(size_t)p * HFDP + o] = ge * a2;               // padded ld (col 255 unused)
}

// zero-pad pout_w (96,255) -> (96,HFDP) so pout GEMM has aligned K=256 loop
__global__ void pack_pout_w(const float* __restrict__ wsrc, float* __restrict__ wdst)
{
    int t = blockIdx.x * blockDim.x + threadIdx.x;
    if (t >= C_IN * HFDP) return;
    int kk = t % HFDP, n = t / HFDP;
    wdst[t] = (kk < HFD) ? wsrc[n * HFD + kk] : 0.f;
}

// final: out NCHW = x2 + ed  (both pixel-major (L,C))
__global__ void final_out(const float* __restrict__ x2, const float* __restrict__ ed,
                          float* __restrict__ out)
{
    int t = blockIdx.x * blockDim.x + threadIdx.x;
    if (t >= L_TOT * C_IN) return;
    int c = t % C_IN, p = t / C_IN;
    out[c * L_TOT + p] = x2[t] + ed[t];
}

// =====================================================================
static inline int cdiv(int a, int b) { return (a + b - 1) / b; }
static inline int gy32(int N) { return cdiv(cdiv(N, 32), 4); }   // grid.y for GEMM

extern "C" void kernel_launch(void* const* d_in, const int* in_sizes, int n_in,
                              void* d_out, int out_size, void* d_ws, size_t ws_size,
                              hipStream_t stream)
{
    const float* x        = (const float*)d_in[0];
    const float* norm1_w  = (const float*)d_in[1];
    const float* norm1_b  = (const float*)d_in[2];
    const float* in_proj  = (const float*)d_in[3];   // (384,96)
    const float* conv_w   = (const float*)d_in[4];   // (192,1,3,3)
    const float* conv_b   = (const float*)d_in[5];
    const float* xproj_w  = (const float*)d_in[6];   // (4,38,192)
    const float* dt_w     = (const float*)d_in[7];   // (4,192,6)
    const float* dt_b     = (const float*)d_in[8];   // (4,192)
    const float* A_logs   = (const float*)d_in[9];   // (768,16)
    const float* Ds       = (const float*)d_in[10];  // (768)
    const float* onw      = (const float*)d_in[11];
    const float* onb      = (const float*)d_in[12];
    const float* oproj_w  = (const float*)d_in[13];  // (96,192)
    const float* norm2_w  = (const float*)d_in[14];
    const float* norm2_b  = (const float*)d_in[15];
    const float* pin_w    = (const float*)d_in[16];  // (510,96)
    const float* dw_w     = (const float*)d_in[17];  // (510,1,3,3)
    const float* pout_w   = (const float*)d_in[18];  // (96,255)
    float* out = (float*)d_out;

    // ---- workspace bump allocator (floats) ----
    float* ws = (float*)d_ws;
    float* n1    = ws; ws += L_TOT * C_IN;
    float* xz    = ws; ws += L_TOT * 2 * DI;
    float* xs    = ws; ws += (size_t)KDIR * L_TOT * DI;
    float* xdbl  = ws; ws += (size_t)KDIR * L_TOT * XC;
    float* delta = ws; ws += (size_t)KDIR * L_TOT * DI;
    float* agg   = ws; ws += (size_t)2 * KDIR * DI * NS * CHUNKS;
    float* hIn   = ws; ws += (size_t)KDIR * DI * NS * CHUNKS;
    float* ysb   = ws; ws += (size_t)KDIR * L_TOT * DI;
    float* yn    = ws; ws += L_TOT * DI;
    float* attn  = ws; ws += L_TOT * C_IN;
    float* x2    = ws; ws += L_TOT * C_IN;
    float* n2    = ws; ws += L_TOT * C_IN;
    float* hbuf  = ws; ws += L_TOT * 2 * HFD;
    float* gbuf  = ws; ws += L_TOT * HFDP;           // padded ld = 256
    float* ed    = ws; ws += L_TOT * C_IN;
    float* poutp = ws; ws += C_IN * HFDP;            // packed pout_w
    (void)ws_size; (void)in_sizes; (void)n_in; (void)out_size;

    const dim3 gb(128, 1, 1);          // 4 wave32 per block (WMMA kernels)
    const int MT = L_TOT / 16;         // 256 M-tiles everywhere

    // 0) pack pout_w once per call (deterministic, tiny)
    pack_pout_w<<<cdiv(C_IN * HFDP, 256), 256, 0, stream>>>(pout_w, poutp);

    // 1) norm1
    rmsln_nchw<<<cdiv(L_TOT, 256), 256, 0, stream>>>(x, norm1_w, norm1_b, n1, C_IN);

    // 2) in_proj: xz(L,384) = n1(L,96) @ W(384,96)^T
    gemm_wmma<0><<<dim3(MT, gy32(2 * DI), 1), gb, 0, stream>>>(
        n1, in_proj, xz, nullptr, L_TOT, 2 * DI, C_IN, C_IN, C_IN, 2 * DI, 0, 0, 0, 0);

    // 3) dwconv + SiLU + 4-direction scatter
    conv_silu_scatter<<<cdiv(L_TOT * DI, 256), 256, 0, stream>>>(xz, conv_w, conv_b, xs);

    // 4) x_dbl (batched over 4 dirs): (L,38) = xs(L,192) @ xproj(38,192)^T
    gemm_wmma<0><<<dim3(MT, gy32(XC), KDIR), gb, 0, stream>>>(
        xs, xproj_w, xdbl, nullptr, L_TOT, XC, DI, DI, DI, XC,
        L_TOT * DI, XC * DI, L_TOT * XC, 0);

    // 5) delta (batched): (L,192) = dts(L,6) @ dt_w(192,6)^T, +bias, softplus
    gemm_wmma<1><<<dim3(MT, gy32(DI), KDIR), gb, 0, stream>>>(
        xdbl, dt_w, delta, dt_b, L_TOT, DI, RK, XC, RK, DI,
        L_TOT * XC, DI * RK, L_TOT * DI, DI);

    // 6-8) selective scan, two-level
    scan_phaseA<<<cdiv(KDIR * CHUNKS * NS * DI, 256), 256, 0, stream>>>(
        delta, xs, xdbl, A_logs, agg);
    scan_phaseB<<<cdiv(KDIR * DI * NS, 256), 256, 0, stream>>>(agg, hIn);
    scan_phaseC<<<cdiv(KDIR * CHUNKS * DI, 256), 256, 0, stream>>>(
        delta, xs, xdbl, A_logs, Ds, hIn, ysb);

    // 9) merge directions + LN + SiLU(z) gate
    combine_ln_gate<<<cdiv(L_TOT, 256), 256, 0, stream>>>(ysb, xz, onw, onb, yn);

    // 10) out_proj: attn(L,96) = yn(L,192) @ W(96,192)^T
    gemm_wmma<0><<<dim3(MT, gy32(C_IN), 1), gb, 0, stream>>>(
        yn, oproj_w, attn, nullptr, L_TOT, C_IN, DI, DI, DI, C_IN, 0, 0, 0, 0);

    // 11) residual
    resid_add<<<cdiv(L_TOT * C_IN, 256), 256, 0, stream>>>(x, attn, x2);

    // 12) norm2
    rmsln_lc<<<cdiv(L_TOT, 256), 256, 0, stream>>>(x2, norm2_w, norm2_b, n2, C_IN);

    // 13) pin: hbuf(L,510) = n2(L,96) @ pin_w(510,96)^T
    gemm_wmma<0><<<dim3(MT, gy32(2 * HFD), 1), gb, 0, stream>>>(
        n2, pin_w, hbuf, nullptr, L_TOT, 2 * HFD, C_IN, C_IN, C_IN, 2 * HFD, 0, 0, 0, 0);

    // 14) dwconv + gelu gate -> gbuf (L,256 padded)
    dwconv_gelugate<<<cdiv(L_TOT * HFD, 256), 256, 0, stream>>>(hbuf, dw_w, gbuf);

    // 15) pout: ed(L,96) = g(L,255) @ pout_w(96,255)^T  (K padded to 256, aligned)
    gemm_wmma<0><<<dim3(MT, gy32(C_IN), 1), gb, 0, stream>>>(
        gbuf, poutp, ed, nullptr, L_TOT, C_IN, HFDP, HFDP, HFDP, C_IN, 0, 0, 0, 0);

    // 16) final residual, back to NCHW
    final_out<<<cdiv(L_TOT * C_IN, 256), 256, 0, stream>>>(x2, ed, out);
}